// MyModel_61933428410985
// MI455X (gfx1250) — compile-verified
//
#include <hip/hip_runtime.h>

// ---------------------------------------------------------------------------
// Device-side persistent L-BFGS (torch.optim.LBFGS port) for MI455X / gfx1250.
// One cooperative kernel runs the whole optimizer; all scalar control flow is
// replicated uniformly across threads from grid-uniform reductions.
// CDNA5 specifics: wave32 all-reduce via V_WMMA_F32_16X16X4_F32, b128
// vectorized passes, global_prefetch of the next history row, software grid
// barrier (one barrier per reduction via double-buffered partials).
// ---------------------------------------------------------------------------

#define NELEM   131072
#define NQ      (NELEM / 4)          // float4 elements per vector
#define HISTN   100
#define NBLK    64
#define BLOCK   256
#define NWAVES  (BLOCK / 32)
#define GSIZE   (NBLK * BLOCK)

#define TOL_GRAD   1e-5
#define TOL_CHANGE 1e-9
#define WOLFE_C1   1e-4
#define WOLFE_C2   0.9
#define MAX_LS     25

typedef float v2f __attribute__((ext_vector_type(2)));
typedef float v4f __attribute__((ext_vector_type(4)));
typedef float v8f __attribute__((ext_vector_type(8)));

struct Ctx {
  float* x; float* grad; float* prevg; float* d; float* q; float* xt;
  float* gnew; float* gprev; float* bg0; float* bg1;
  float* S; float* Y;
  float* partials;            // 2*NBLK floats (double buffered)
  unsigned* cnt; unsigned* gen;
  int gtid;
  int redcnt;                 // uniform reduction counter (buffer parity)
};

__device__ inline v4f ld4(const float* p, int i4) { return ((const v4f*)p)[i4]; }
__device__ inline void st4(float* p, int i4, v4f v) { ((v4f*)p)[i4] = v; }
__device__ inline float dot4(v4f a, v4f b, float p) {
  p = fmaf(a[0], b[0], p); p = fmaf(a[1], b[1], p);
  p = fmaf(a[2], b[2], p); p = fmaf(a[3], b[3], p);
  return p;
}

// ---------------- grid barrier (sense-reversing, in workspace) -------------
__device__ inline void gsync(Ctx& c) {
  __threadfence();
  __syncthreads();
  if (threadIdx.x == 0) {
    volatile unsigned* vgen = (volatile unsigned*)c.gen;
    unsigned g = *vgen;
    unsigned prev = atomicAdd(c.cnt, 1u);
    if (prev == (unsigned)(NBLK - 1)) {
      *((volatile unsigned*)c.cnt) = 0u;
      __threadfence();
      atomicAdd(c.gen, 1u);
    } else {
      while (*vgen == g) { __builtin_amdgcn_s_sleep(1); }
    }
  }
  __syncthreads();
}

// ---------------- wave32 all-reduce sum via WMMA ---------------------------
// A (16x4 f32): vgpr0 = lane partial, vgpr1 = 0 => A[M][0]=p[M], A[M][2]=p[M+16]
// B (4x16) = ones => C[M][N] = p[M] + p[M+16]. Summing the 8 C vgprs gives each
// lane-half the sum of 8 rows; one shfl_xor(16) finishes the 32-lane reduce.
__device__ inline float wave_sum(float v) {
  v2f a; a[0] = v;    a[1] = 0.0f;
  v2f b; b[0] = 1.0f; b[1] = 1.0f;
  v8f acc = {};
  acc = __builtin_amdgcn_wmma_f32_16x16x4_f32(false, a, false, b,
                                              (short)0, acc, false, false);
  float s = acc[0] + acc[1] + acc[2] + acc[3] +
            acc[4] + acc[5] + acc[6] + acc[7];
  s += __shfl_xor(s, 16, 32);
  return s;  // identical in every lane
}

__device__ inline float wave_max(float v) {
  for (int off = 16; off > 0; off >>= 1)
    v = fmaxf(v, __shfl_xor(v, off, 32));
  return v;
}

// ---------------- grid-wide reductions (one barrier, double-buffered) ------
__device__ float gsum(Ctx& c, float p) {
  __shared__ float sm[NWAVES];
  float w = wave_sum(p);
  int lane = threadIdx.x & 31, wid = threadIdx.x >> 5;
  if (lane == 0) sm[wid] = w;
  __syncthreads();
  float blk = 0.0f;
  for (int k = 0; k < NWAVES; k++) blk += sm[k];
  __syncthreads();
  float* buf = c.partials + (c.redcnt & 1) * NBLK;
  c.redcnt++;
  if (threadIdx.x == 0) buf[blockIdx.x] = blk;
  gsync(c);
  float tot = 0.0f;
  for (int k = 0; k < NBLK; k++) tot += buf[k];
  return tot;  // safe: this buffer is next written two reductions from now
}

__device__ float gmaxr(Ctx& c, float p) {
  __shared__ float sm[NWAVES];
  float w = wave_max(p);
  int lane = threadIdx.x & 31, wid = threadIdx.x >> 5;
  if (lane == 0) sm[wid] = w;
  __syncthreads();
  float blk = -3.402823466e38f;
  for (int k = 0; k < NWAVES; k++) blk = fmaxf(blk, sm[k]);
  __syncthreads();
  float* buf = c.partials + (c.redcnt & 1) * NBLK;
  c.redcnt++;
  if (threadIdx.x == 0) buf[blockIdx.x] = blk;
  gsync(c);
  float tot = -3.402823466e38f;
  for (int k = 0; k < NBLK; k++) tot = fmaxf(tot, buf[k]);
  return tot;
}

// ---------------- vectorized vector primitives -----------------------------
__device__ float gmaxabs(Ctx& c, const float* a) {
  float p = 0.0f;
  for (int i4 = c.gtid; i4 < NQ; i4 += GSIZE) {
    v4f v = ld4(a, i4);
    p = fmaxf(p, fmaxf(fmaxf(fabsf(v[0]), fabsf(v[1])),
                       fmaxf(fabsf(v[2]), fabsf(v[3]))));
  }
  return gmaxr(c, p);
}

__device__ float gsumabs(Ctx& c, const float* a) {
  float p = 0.0f;
  for (int i4 = c.gtid; i4 < NQ; i4 += GSIZE) {
    v4f v = ld4(a, i4);
    p += fabsf(v[0]) + fabsf(v[1]) + fabsf(v[2]) + fabsf(v[3]);
  }
  return gsum(c, p);
}

__device__ void vcopy(Ctx& c, float* dst, const float* s) {
  for (int i4 = c.gtid; i4 < NQ; i4 += GSIZE) st4(dst, i4, ld4(s, i4));
  gsync(c);
}

// dst = a*u + v (dst may alias v); if nd: returns grid dot(nd, dst_new);
// pfrow: prefetch of the row needed by the NEXT fused pass.
__device__ double fused_axpy_dot(Ctx& c, float* dst, float a, const float* u,
                                 const float* v, const float* nd,
                                 const float* pfrow) {
  float p = 0.0f;
  for (int i4 = c.gtid; i4 < NQ; i4 += GSIZE) {
    if (pfrow) __builtin_prefetch((const void*)(pfrow + 4 * i4), 0, 1);
    v4f uv = ld4(u, i4), vv = ld4(v, i4);
    v4f r;
    r[0] = fmaf(a, uv[0], vv[0]); r[1] = fmaf(a, uv[1], vv[1]);
    r[2] = fmaf(a, uv[2], vv[2]); r[3] = fmaf(a, uv[3], vv[3]);
    st4(dst, i4, r);
    if (nd) p = dot4(ld4(nd, i4), r, p);
  }
  if (nd) return (double)gsum(c, p);
  gsync(c);
  return 0.0;
}

// ---------------- Rosenbrock value + grad (+ fused grad.dot(dv)) -----------
__device__ void eval_fg_dot(Ctx& c, const float* xv, float* g, const float* dv,
                            double& f_out, double& gtd_out, bool want_dot) {
  float pf_ = 0.0f, pd = 0.0f;
  for (int i4 = c.gtid; i4 < NQ; i4 += GSIZE) {
    int i = 4 * i4;
    v4f x4 = ld4(xv, i4);
    float xnext = (i + 4 < NELEM) ? xv[i + 4] : 0.0f;
    float xprev = (i > 0) ? xv[i - 1] : 0.0f;
    v4f g4;
#pragma unroll
    for (int k = 0; k < 4; k++) {
      float xi = x4[k];
      float xp = (k < 3) ? x4[k + 1] : xnext;
      float xm = (k > 0) ? x4[k - 1] : xprev;
      float gi = 0.0f;
      if (i + k < NELEM - 1) {
        float t1 = xp - xi * xi;
        float t2 = 1.0f - xi;
        pf_ += 100.0f * t1 * t1 + t2 * t2;
        gi += -400.0f * xi * t1 - 2.0f * t2;
      }
      if (i + k > 0) gi += 200.0f * (xi - xm * xm);
      g4[k] = gi;
    }
    st4(g, i4, g4);
    if (want_dot) pd = dot4(g4, ld4(dv, i4), pd);
  }
  f_out = (double)gsum(c, pf_);   // gsync inside publishes g too
  if (want_dot) gtd_out = (double)gsum(c, pd);
}

// ---------------- cubic interpolation (scalar, double) ---------------------
__device__ double cubic_interp(double x1, double f1, double g1,
                               double x2, double f2, double g2,
                               bool hasB, double bmin, double bmax) {
  double xmin_b, xmax_b;
  if (hasB) { xmin_b = bmin; xmax_b = bmax; }
  else if (x1 <= x2) { xmin_b = x1; xmax_b = x2; }
  else { xmin_b = x2; xmax_b = x1; }
  double d1 = g1 + g2 - 3.0 * (f1 - f2) / (x1 - x2);
  double d2s = d1 * d1 - g1 * g2;
  if (d2s >= 0.0) {
    double d2 = sqrt(d2s);
    double mp;
    if (x1 <= x2) mp = x2 - (x2 - x1) * ((g2 + d2 - d1) / (g2 - g1 + 2.0 * d2));
    else          mp = x1 - (x1 - x2) * ((g1 + d2 - d1) / (g1 - g2 + 2.0 * d2));
    return fmin(fmax(mp, xmin_b), xmax_b);
  }
  return (xmin_b + xmax_b) / 2.0;
}

// ---------------- strong Wolfe line search (torch port) --------------------
__device__ int strong_wolfe(Ctx& c, double& t, double f, double gtd,
                            double& floss) {
  double d_norm = (double)gmaxabs(c, c.d);

  auto obj = [&](double tt, double& fv) -> double {
    fused_axpy_dot(c, c.xt, (float)tt, c.d, c.x, nullptr, nullptr); // xt=x+t*d
    double gtd_n;
    eval_fg_dot(c, c.xt, c.gnew, c.d, fv, gtd_n, true);
    return gtd_n;
  };

  double f_new;
  double gtd_new = obj(t, f_new);
  int evals = 1;
  double t_prev = 0.0, f_prev = f, gtd_prev = gtd;
  vcopy(c, c.gprev, c.grad);

  bool done = false, have_br = false;
  int ls_iter = 0, nbr = 0;
  double br_t[2], br_f[2], br_g[2];
  float* bg[2] = { c.bg0, c.bg1 };

  while (ls_iter < MAX_LS) {
    bool armijo = (f_new > (f + WOLFE_C1 * t * gtd)) ||
                  (ls_iter > 1 && f_new >= f_prev);
    if (armijo) {
      br_t[0] = t_prev; br_t[1] = t;
      br_f[0] = f_prev; br_f[1] = f_new;
      br_g[0] = gtd_prev; br_g[1] = gtd_new;
      vcopy(c, bg[0], c.gprev); vcopy(c, bg[1], c.gnew);
      nbr = 2; have_br = true; break;
    }
    if (fabs(gtd_new) <= -WOLFE_C2 * gtd) {
      br_t[0] = t; br_f[0] = f_new; br_g[0] = gtd_new;
      vcopy(c, bg[0], c.gnew);
      nbr = 1; have_br = true; done = true; break;
    }
    if (gtd_new >= 0.0) {
      br_t[0] = t_prev; br_t[1] = t;
      br_f[0] = f_prev; br_f[1] = f_new;
      br_g[0] = gtd_prev; br_g[1] = gtd_new;
      vcopy(c, bg[0], c.gprev); vcopy(c, bg[1], c.gnew);
      nbr = 2; have_br = true; break;
    }
    double min_step = t + 0.01 * (t - t_prev);
    double max_step = t * 10.0;
    double tmp = t;
    t = cubic_interp(t_prev, f_prev, gtd_prev, t, f_new, gtd_new,
                     true, min_step, max_step);
    t_prev = tmp; f_prev = f_new; gtd_prev = gtd_new;
    vcopy(c, c.gprev, c.gnew);
    gtd_new = obj(t, f_new);
    evals++; ls_iter++;
  }
  if (!have_br) {
    br_t[0] = 0.0; br_t[1] = t;
    br_f[0] = f;   br_f[1] = f_new;
    br_g[0] = gtd; br_g[1] = gtd_new;
    vcopy(c, bg[0], c.grad); vcopy(c, bg[1], c.gnew);
    nbr = 2;
  }

  int last = nbr - 1;
  bool insuf = false;
  int low = (br_f[0] <= br_f[last]) ? 0 : 1;
  int high = 1 - low;

  while (!done && ls_iter < MAX_LS) {
    if (fabs(br_t[1] - br_t[0]) * d_norm < TOL_CHANGE) break;
    t = cubic_interp(br_t[0], br_f[0], br_g[0],
                     br_t[1], br_f[1], br_g[1], false, 0.0, 0.0);
    double mx = fmax(br_t[0], br_t[1]);
    double mn = fmin(br_t[0], br_t[1]);
    double eps = 0.1 * (mx - mn);
    if (fmin(mx - t, t - mn) < eps) {
      if (insuf || t >= mx || t <= mn) {
        t = (fabs(t - mx) < fabs(t - mn)) ? (mx - eps) : (mn + eps);
        insuf = false;
      } else insuf = true;
    } else insuf = false;

    gtd_new = obj(t, f_new);
    evals++; ls_iter++;

    if (f_new > (f + WOLFE_C1 * t * gtd) || f_new >= br_f[low]) {
      br_t[high] = t; br_f[high] = f_new; br_g[high] = gtd_new;
      vcopy(c, bg[high], c.gnew);
      low = (br_f[0] <= br_f[1]) ? 0 : 1; high = 1 - low;
    } else {
      if (fabs(gtd_new) <= -WOLFE_C2 * gtd) {
        done = true;
      } else if (gtd_new * (br_t[high] - br_t[low]) >= 0.0) {
        br_t[high] = br_t[low]; br_f[high] = br_f[low]; br_g[high] = br_g[low];
        vcopy(c, bg[high], bg[low]);
      }
      br_t[low] = t; br_f[low] = f_new; br_g[low] = gtd_new;
      vcopy(c, bg[low], c.gnew);
    }
  }

  t = br_t[low];
  floss = br_f[low];
  vcopy(c, c.grad, bg[low]);
  return evals;
}

// ---------------- init kernel: zero barrier words (ws is poisoned) ---------
__global__ void lbfgs_init_sync(unsigned* cnt, unsigned* gen) {
  *cnt = 0u; *gen = 0u;
}

// ---------------- main persistent kernel -----------------------------------
__global__ __launch_bounds__(BLOCK, 1)
void lbfgs_kernel(const float* __restrict__ x0, float* __restrict__ out,
                  float* __restrict__ w, unsigned* cnt, unsigned* gen) {
  Ctx c;
  c.x     = w;
  c.grad  = w + 1L * NELEM;
  c.prevg = w + 2L * NELEM;
  c.d     = w + 3L * NELEM;
  c.q     = w + 4L * NELEM;
  c.xt    = w + 5L * NELEM;
  c.gnew  = w + 6L * NELEM;
  c.gprev = w + 7L * NELEM;
  c.bg0   = w + 8L * NELEM;
  c.bg1   = w + 9L * NELEM;
  c.S     = w + 10L * NELEM;
  c.Y     = w + (10L + HISTN) * NELEM;
  c.partials = w + (10L + 2L * HISTN) * NELEM;
  c.cnt = cnt; c.gen = gen;
  c.gtid = (int)(blockIdx.x * BLOCK + threadIdx.x);
  c.redcnt = 0;

  double rho[HISTN];  // indexed by physical slot (grid-uniform)
  double al[HISTN];   // indexed by logical history position

  for (int run = 0; run < 2; run++) {
    const int max_iter = (run == 0) ? 1000 : 100;
    const int nsteps   = (run == 0) ? 1 : 10;
    const int max_eval = max_iter * 5 / 4;

    for (int i4 = c.gtid; i4 < NQ; i4 += GSIZE) st4(c.x, i4, ld4(x0, i4));
    gsync(c);

    int hcount = 0, hstart = 0;
    long n_iter_total = 0;
    double H_diag = 1.0, tstep = 0.0, loss = 0.0, prev_loss = 0.0;

    for (int sc = 0; sc < nsteps; sc++) {
      double dummy;
      eval_fg_dot(c, c.x, c.grad, nullptr, loss, dummy, false);
      if ((double)gmaxabs(c, c.grad) <= TOL_GRAD) continue;

      int current_evals = 1;
      int n_iter = 0;
      while (n_iter < max_iter && current_evals < max_eval) {
        n_iter++; n_iter_total++;

        if (n_iter_total == 1) {
          for (int i4 = c.gtid; i4 < NQ; i4 += GSIZE) {
            v4f g4 = ld4(c.grad, i4);
            v4f d4; d4[0] = -g4[0]; d4[1] = -g4[1]; d4[2] = -g4[2]; d4[3] = -g4[3];
            st4(c.d, i4, d4);
          }
          gsync(c);
          hcount = 0; hstart = 0;
        } else {
          // fused: y=g-gprev, s=t*d, ys=y.s, yy=y.y (one data pass)
          float ts = (float)tstep;
          float p1 = 0.0f, p2 = 0.0f;
          for (int i4 = c.gtid; i4 < NQ; i4 += GSIZE) {
            v4f gv = ld4(c.grad, i4), pg = ld4(c.prevg, i4), dv = ld4(c.d, i4);
            v4f y4; y4[0] = gv[0] - pg[0]; y4[1] = gv[1] - pg[1];
                    y4[2] = gv[2] - pg[2]; y4[3] = gv[3] - pg[3];
            v4f s4; s4[0] = ts * dv[0]; s4[1] = ts * dv[1];
                    s4[2] = ts * dv[2]; s4[3] = ts * dv[3];
            st4(c.gnew, i4, y4); st4(c.xt, i4, s4);
            p1 = dot4(y4, s4, p1);
            p2 = dot4(y4, y4, p2);
          }
          double ys = (double)gsum(c, p1);
          double yy = (double)gsum(c, p2);
          if (ys > 1e-10) {
            int phys;
            if (hcount == HISTN) { phys = hstart; hstart = (hstart + 1) % HISTN; }
            else { phys = (hstart + hcount) % HISTN; hcount++; }
            float* Srow = c.S + (long)phys * NELEM;
            float* Yrow = c.Y + (long)phys * NELEM;
            for (int i4 = c.gtid; i4 < NQ; i4 += GSIZE) {  // fused S,Y store
              st4(Srow, i4, ld4(c.xt, i4));
              st4(Yrow, i4, ld4(c.gnew, i4));
            }
            gsync(c);
            rho[phys] = 1.0 / ys;
            H_diag = ys / yy;
          }

          // two-loop recursion, fused: each pass does axpy_j + dot_{j-1}
          const float* Slast = (hcount > 0)
              ? c.S + (long)((hstart + hcount - 1) % HISTN) * NELEM : nullptr;
          float p = 0.0f;
          for (int i4 = c.gtid; i4 < NQ; i4 += GSIZE) {  // q=-g (+ first dot)
            v4f gv = ld4(c.grad, i4);
            v4f qv; qv[0] = -gv[0]; qv[1] = -gv[1]; qv[2] = -gv[2]; qv[3] = -gv[3];
            st4(c.q, i4, qv);
            if (Slast) p = dot4(ld4(Slast, i4), qv, p);
          }
          double dotv = 0.0;
          if (Slast) dotv = (double)gsum(c, p); else gsync(c);

          for (int j = hcount - 1; j >= 0; j--) {
            int ph = (hstart + j) % HISTN;
            const float* Yr = c.Y + (long)ph * NELEM;
            double a = rho[ph] * dotv;
            al[j] = a;
            const float* nd  = (j > 0)
                ? c.S + (long)((hstart + j - 1) % HISTN) * NELEM : nullptr;
            const float* pfr = (j > 0)
                ? c.Y + (long)((hstart + j - 1) % HISTN) * NELEM : nullptr;
            dotv = fused_axpy_dot(c, c.q, -(float)a, Yr, c.q, nd, pfr);
          }

          const float* Y0 = (hcount > 0)
              ? c.Y + (long)(hstart % HISTN) * NELEM : nullptr;
          float hd = (float)H_diag;
          p = 0.0f;
          for (int i4 = c.gtid; i4 < NQ; i4 += GSIZE) {  // d=H_diag*q (+dot)
            v4f qv = ld4(c.q, i4);
            v4f dv; dv[0] = hd * qv[0]; dv[1] = hd * qv[1];
                    dv[2] = hd * qv[2]; dv[3] = hd * qv[3];
            st4(c.d, i4, dv);
            if (Y0) p = dot4(ld4(Y0, i4), dv, p);
          }
          if (Y0) dotv = (double)gsum(c, p); else gsync(c);

          for (int j = 0; j < hcount; j++) {
            int ph = (hstart + j) % HISTN;
            const float* Sr = c.S + (long)ph * NELEM;
            double b = rho[ph] * dotv;
            float coef = (float)(al[j] - b);
            const float* nd  = (j + 1 < hcount)
                ? c.Y + (long)((hstart + j + 1) % HISTN) * NELEM : nullptr;
            const float* pfr = (j + 1 < hcount)
                ? c.S + (long)((hstart + j + 1) % HISTN) * NELEM : nullptr;
            dotv = fused_axpy_dot(c, c.d, coef, Sr, c.d, nd, pfr);
          }
        }

        // fused: prev_grad = grad, gtd = grad.d (one data pass)
        double gtd;
        {
          float p = 0.0f;
          for (int i4 = c.gtid; i4 < NQ; i4 += GSIZE) {
            v4f gv = ld4(c.grad, i4);
            st4(c.prevg, i4, gv);
            p = dot4(gv, ld4(c.d, i4), p);
          }
          gtd = (double)gsum(c, p);
        }
        prev_loss = loss;

        if (n_iter_total == 1) {
          double s1 = (double)gsumabs(c, c.grad);
          tstep = fmin(1.0, 1.0 / s1);
        } else {
          tstep = 1.0;
        }

        if (gtd > -TOL_CHANGE) break;

        int ls_evals = strong_wolfe(c, tstep, loss, gtd, loss);

        float ts2 = (float)tstep;
        for (int i4 = c.gtid; i4 < NQ; i4 += GSIZE) {   // x += t*d
          v4f xv = ld4(c.x, i4), dv = ld4(c.d, i4);
          v4f r; r[0] = fmaf(ts2, dv[0], xv[0]); r[1] = fmaf(ts2, dv[1], xv[1]);
                 r[2] = fmaf(ts2, dv[2], xv[2]); r[3] = fmaf(ts2, dv[3], xv[3]);
          st4(c.x, i4, r);
        }
        gsync(c);

        current_evals += ls_evals;
        if (n_iter == max_iter) break;
        if (current_evals >= max_eval) break;
        if ((double)gmaxabs(c, c.grad) <= TOL_GRAD) break;
        if (tstep * (double)gmaxabs(c, c.d) <= TOL_CHANGE) break;
        if (fabs(loss - prev_loss) < TOL_CHANGE) break;
      }
    }

    // convergence flag: ||x - 1||_2 < 1e-4
    float p = 0.0f;
    for (int i4 = c.gtid; i4 < NQ; i4 += GSIZE) {
      v4f xv = ld4(c.x, i4);
      v4f t; t[0] = xv[0] - 1.0f; t[1] = xv[1] - 1.0f;
             t[2] = xv[2] - 1.0f; t[3] = xv[3] - 1.0f;
      p = dot4(t, t, p);
    }
    double nn = sqrt((double)gsum(c, p));
    if (c.gtid == 0) out[run] = (nn < 1e-4) ? 1.0f : 0.0f;
    gsync(c);
  }
}

// ---------------------------------------------------------------------------
extern "C" void kernel_launch(void* const* d_in, const int* in_sizes, int n_in,
                              void* d_out, int out_size, void* d_ws, size_t ws_size,
                              hipStream_t stream) {
  (void)in_sizes; (void)n_in; (void)out_size; (void)ws_size;
  const float* x0 = (const float*)d_in[0];
  float* out = (float*)d_out;
  float* w = (float*)d_ws;
  // layout: 10 N-vectors | S (100xN) | Y (100xN) | partials[2*NBLK] | cnt,gen
  unsigned* cnt = (unsigned*)(w + (10L + 2L * HISTN) * NELEM + 2L * NBLK);
  unsigned* gen = cnt + 1;

  lbfgs_init_sync<<<1, 1, 0, stream>>>(cnt, gen);
  lbfgs_kernel<<<dim3(NBLK), dim3(BLOCK), 0, stream>>>(x0, out, w, cnt, gen);
}